// ElasticTransform_8280696946889
// MI455X (gfx1250) — compile-verified
//
#include <hip/hip_runtime.h>
#include <hip/hip_bf16.h>
#include <math.h>

typedef __attribute__((ext_vector_type(2))) float v2f;
typedef __attribute__((ext_vector_type(8))) float v8f;

#define KS    63
#define KPAD  31        // KS/2
#define KEXT  80        // 78 (=16+62) padded up to multiple of 4
#define AST   84        // LDS row stride for horizontal A tile (conflict-free)
#define NSTEP 20        // KEXT/4 WMMA steps

// Problem dimensions (fixed by the reference).
#define DB 16
#define DC 3
#define DH 512
#define DW 512
#define DHW (DH * DW)

// ---------------------------------------------------------------------------
// Kernel 0: packed Toeplitz weight fragments in WMMA lane layout.
// pack[(s*32 + lane)] = v2f{ w[k0-idx], w[k0+1-idx] },
// k0 = 4s + 2*(lane>>4), idx = lane&15, w = normalized 63-tap Gaussian,
// zero outside [0,63). Same table serves hblur's B and vblur's A operand.
// ---------------------------------------------------------------------------
__global__ void et_wpack(float* __restrict__ pack) {
    int tid = threadIdx.x;                 // 640 threads = 20 steps * 32 lanes
    const float inv2s2 = 1.0f / 2048.0f;   // 1/(2*sigma^2), sigma=32
    float sum = 0.0f;
    for (int i = 0; i < KS; ++i) {
        float d = (float)(i - KPAD);
        sum += expf(-d * d * inv2s2);
    }
    int s    = tid >> 5;
    int lane = tid & 31;
    int k0   = (s << 2) + ((lane >> 4) << 1);
    int idx  = lane & 15;
    int t0   = k0 - idx;
    float v0 = 0.0f, v1 = 0.0f;
    if ((unsigned)t0 < (unsigned)KS) {
        float d = (float)(t0 - KPAD);
        v0 = expf(-d * d * inv2s2) / sum;
    }
    if ((unsigned)(t0 + 1) < (unsigned)KS) {
        float d = (float)(t0 + 1 - KPAD);
        v1 = expf(-d * d * inv2s2) / sum;
    }
    pack[tid * 2 + 0] = v0;
    pack[tid * 2 + 1] = v1;
}

// ---------------------------------------------------------------------------
// Kernel 1: horizontal 63-tap blur of (noise*2-1) via WMMA f32 16x16x4.
// One 16x16 output tile per wave. out[m][n] = sum_j A[m][j]*B[j][n],
// A[m][j] = x[row0+m][col0+j-31] (zero pad), B = Toeplitz (preloaded regs).
// ---------------------------------------------------------------------------
__global__ void __launch_bounds__(256)
et_hblur(const float* __restrict__ noise, float* __restrict__ tmp,
         const float* __restrict__ wpack) {
    __shared__ float at[8][16 * AST];

    const int tid  = threadIdx.x;
    const int wave = tid >> 5;
    const int lane = tid & 31;

    constexpr int tiles_x = DW >> 4;                  // 32
    constexpr int tiles_per_img = tiles_x * (DH >> 4);// 1024
    const int tile = blockIdx.x * 8 + wave;
    const int bc   = tile / tiles_per_img;            // [0, 2B)
    const int rem  = tile - bc * tiles_per_img;
    const int row0 = (rem / tiles_x) << 4;
    const int col0 = (rem % tiles_x) << 4;

    // Preload Toeplitz fragments into registers (coalesced, L2-resident).
    const v2f* wp = (const v2f*)wpack;
    v2f wreg[NSTEP];
    #pragma unroll
    for (int s = 0; s < NSTEP; ++s) wreg[s] = wp[(s << 5) + lane];

    const float* src = noise + (size_t)bc * DHW;
    float* a = at[wave];

    // Stage 16 rows x 80 cols (cols col0-31 .. col0+48), zero pad, 2x-1.
    #pragma unroll 4
    for (int i = 0; i < 40; ++i) {
        int e  = lane + (i << 5);       // 0..1279
        int rr = e / KEXT;
        int cc = e - rr * KEXT;
        int c  = col0 + cc - KPAD;
        float v = 0.0f;
        if (c >= 0 && c < DW)
            v = src[rr * DW + (row0 * DW + c)] * 2.0f - 1.0f;
        a[rr * AST + cc] = v;
    }
    __syncthreads();

    const int n    = lane & 15;
    const int half = lane >> 4;            // 0: K pair {0,1}, 1: {2,3}
    const int abase = n * AST + (half << 1);
    v8f c = {};
    #pragma unroll
    for (int s = 0; s < NSTEP; ++s) {
        v2f A;
        A.x = a[abase + (s << 2)];
        A.y = a[abase + (s << 2) + 1];
        c = __builtin_amdgcn_wmma_f32_16x16x4_f32(
                false, A, false, wreg[s], (short)0, c, false, false);
    }

    // Single base address; row offsets are compile-time (r*DW*4 = 2048r B).
    float* dst = tmp + (size_t)bc * DHW + (row0 + (half << 3)) * DW + col0 + n;
    #pragma unroll
    for (int r = 0; r < 8; ++r) dst[r * DW] = c[r];
}

// ---------------------------------------------------------------------------
// Kernel 2: vertical 63-tap blur; Toeplitz in A (preloaded regs), image
// columns in B. Writes disp interleaved [B,H,W,2].
// ---------------------------------------------------------------------------
__global__ void __launch_bounds__(256)
et_vblur(const float* __restrict__ tmp, float* __restrict__ disp,
         const float* __restrict__ wpack) {
    __shared__ float bt[8][KEXT * 16];

    const int tid  = threadIdx.x;
    const int wave = tid >> 5;
    const int lane = tid & 31;

    constexpr int tiles_x = DW >> 4;
    constexpr int tiles_per_img = tiles_x * (DH >> 4);
    const int tile = blockIdx.x * 8 + wave;
    const int bc   = tile / tiles_per_img;            // [0, 2B)
    const int rem  = tile - bc * tiles_per_img;
    const int row0 = (rem / tiles_x) << 4;
    const int col0 = (rem % tiles_x) << 4;
    const int b    = bc >> 1;
    const int ch   = bc & 1;

    const v2f* wp = (const v2f*)wpack;
    v2f wreg[NSTEP];
    #pragma unroll
    for (int s = 0; s < NSTEP; ++s) wreg[s] = wp[(s << 5) + lane];

    const float* src = tmp + (size_t)bc * DHW;
    float* btile = bt[wave];

    // Stage 80 rows x 16 cols (rows row0-31 .. row0+48), zero pad.
    #pragma unroll 4
    for (int i = 0; i < 40; ++i) {
        int e  = lane + (i << 5);       // 0..1279
        int rr = e >> 4;                // 0..79
        int cc = e & 15;
        int gr = row0 + rr - KPAD;
        float v = 0.0f;
        if (gr >= 0 && gr < DH)
            v = src[gr * DW + col0 + cc];
        btile[(rr << 4) + cc] = v;
    }
    __syncthreads();

    const int n    = lane & 15;
    const int half = lane >> 4;
    const int bbase = (half << 5) + n;     // (2*half)*16 + n
    v8f c = {};
    #pragma unroll
    for (int s = 0; s < NSTEP; ++s) {
        v2f Bv;
        Bv.x = btile[bbase + (s << 6)];        // row k0   = 4s+2h
        Bv.y = btile[bbase + (s << 6) + 16];   // row k0+1
        c = __builtin_amdgcn_wmma_f32_16x16x4_f32(
                false, wreg[s], false, Bv, (short)0, c, false, false);
    }

    // Base address + compile-time row offsets (r*2*DW*4 = 4096r bytes).
    float* dst = disp
        + (((size_t)(b * DH + row0 + (half << 3)) * DW + col0 + n) << 1) + ch;
    #pragma unroll
    for (int r = 0; r < 8; ++r) dst[r * 2 * DW] = c[r];
}

// ---------------------------------------------------------------------------
// Kernel 3: bilinear grid_sample (zeros padding, align_corners=False).
// ---------------------------------------------------------------------------
__global__ void __launch_bounds__(256)
et_warp(const float* __restrict__ input, const float* __restrict__ disp,
        float* __restrict__ warped) {
    int idx = blockIdx.x * blockDim.x + threadIdx.x;   // < DB*DHW
    int b = idx >> 18;                 // / DHW (2^18)
    int p = idx & (DHW - 1);
    int y = p >> 9;                    // / DW
    int x = p & (DW - 1);

    float dx = disp[(size_t)idx * 2 + 0];
    float dy = disp[(size_t)idx * 2 + 1];
    float gx = -1.0f + (float)x * (2.0f / (float)(DW - 1));
    float gy = -1.0f + (float)y * (2.0f / (float)(DH - 1));
    float sgx = fminf(fmaxf(gx + dx, -1.0f), 1.0f);
    float sgy = fminf(fmaxf(gy + dy, -1.0f), 1.0f);

    float ix = ((sgx + 1.0f) * (float)DW - 1.0f) * 0.5f;
    float iy = ((sgy + 1.0f) * (float)DH - 1.0f) * 0.5f;
    float ix0 = floorf(ix), iy0 = floorf(iy);
    float wx = ix - ix0, wy = iy - iy0;

    const float* img = input + (size_t)b * DC * DHW;
    float acc0 = 0.0f, acc1 = 0.0f, acc2 = 0.0f;

    float cx[4] = { ix0, ix0 + 1.0f, ix0,        ix0 + 1.0f };
    float cy[4] = { iy0, iy0,        iy0 + 1.0f, iy0 + 1.0f };
    float cw[4] = { (1.0f - wx) * (1.0f - wy), wx * (1.0f - wy),
                    (1.0f - wx) * wy,          wx * wy };
    #pragma unroll
    for (int k = 0; k < 4; ++k) {
        float xf = cx[k], yf = cy[k], wgt = cw[k];
        if (xf >= 0.0f && xf <= (float)(DW - 1) &&
            yf >= 0.0f && yf <= (float)(DH - 1)) {
            int base = ((int)yf << 9) + (int)xf;
            acc0 += img[base]           * wgt;
            acc1 += img[base + DHW]     * wgt;
            acc2 += img[base + 2 * DHW] * wgt;
        }
    }
    float* o = warped + (size_t)b * DC * DHW + p;
    o[0]       = acc0;
    o[DHW]     = acc1;
    o[2 * DHW] = acc2;
}

// ---------------------------------------------------------------------------
extern "C" void kernel_launch(void* const* d_in, const int* in_sizes, int n_in,
                              void* d_out, int out_size, void* d_ws, size_t ws_size,
                              hipStream_t stream) {
    const float* input = (const float*)d_in[0];
    const float* noise = (const float*)d_in[1];
    float* out = (float*)d_out;

    float* wpack = (float*)d_ws;                       // 1280 floats (5KB)
    float* tmp   = (float*)((char*)d_ws + 8192);       // 2B*H*W floats
    float* warped = out;
    float* disp   = out + (size_t)DB * DC * DHW;       // [B,H,W,2]

    et_wpack<<<1, NSTEP * 32, 0, stream>>>(wpack);

    const int tiles = DB * 2 * (DH / 16) * (DW / 16);  // 32768, 8 waves/block
    et_hblur<<<tiles / 8, 256, 0, stream>>>(noise, tmp, wpack);
    et_vblur<<<tiles / 8, 256, 0, stream>>>(tmp, disp, wpack);

    const int npix = DB * DHW;                          // divides 256 exactly
    et_warp<<<npix / 256, 256, 0, stream>>>(input, disp, warped);
}